// VolumeRendering_89790586290719
// MI455X (gfx1250) — compile-verified
//
#include <hip/hip_runtime.h>
#include <hip/hip_bf16.h>

// -----------------------------------------------------------------------------
// Kernel 1: one thread per ray.
//  - binary-search segment bounds in the sorted ray_id array
//  - exact within-ray exclusive log-cumsum -> transmittance, bg_transmittance
//  - register-accumulated per-ray reductions -> integrated_3d/1d, sum_per_ray
// -----------------------------------------------------------------------------
__global__ void __launch_bounds__(256)
vr_ray_scan(const float* __restrict__ alpha,
            const float* __restrict__ v3,
            const float* __restrict__ v1,
            const float* __restrict__ w,
            const float* __restrict__ sv,
            const int*   __restrict__ rid,
            int n, int n_rays,
            float* __restrict__ trans,   // [N]
            float* __restrict__ bg,      // [R]
            float* __restrict__ i3,      // [R,3]
            float* __restrict__ i1,      // [R]
            float* __restrict__ spr)     // [R,3]
{
    const int r = blockIdx.x * blockDim.x + threadIdx.x;
    if (r >= n_rays) return;

    // lower_bound(rid, r)
    int lo = 0, hi = n;
    while (lo < hi) { int m = (lo + hi) >> 1; if (rid[m] <  r) lo = m + 1; else hi = m; }
    const int s = lo;
    // upper_bound(rid, r)  (start from s)
    hi = n;
    while (lo < hi) { int m = (lo + hi) >> 1; if (rid[m] <= r) lo = m + 1; else hi = m; }
    const int e = lo;

    float run = 0.0f;                        // within-ray inclusive log-sum so far
    float a3x = 0.f, a3y = 0.f, a3z = 0.f;   // integrated_3d accumulators
    float a1  = 0.f;                         // integrated_1d accumulator
    float sx  = 0.f, sy = 0.f, sz = 0.f;     // sum_per_ray accumulators

    for (int i = s; i < e; ++i) {
        // Prefetch one 128B line ahead on each stream (gfx1250 global_prefetch_b8).
        if (((i - s) & 31) == 0) {
            int p = i + 32;
            if (p < n) {
                __builtin_prefetch(alpha + p,     0, 3);
                __builtin_prefetch(w     + p,     0, 3);
                __builtin_prefetch(v1    + p,     0, 3);
                __builtin_prefetch(v3    + 3 * p, 0, 3);
                __builtin_prefetch(sv    + 3 * p, 0, 3);
            }
        }
        const float a  = alpha[i];
        const float wi = w[i];
        const float l  = logf(fminf(fmaxf(1.0f - a, 1e-7f), 1.0f));
        trans[i] = expf(run);                // exclusive: before adding this sample
        run += l;

        a3x += v3[3 * i + 0] * wi;
        a3y += v3[3 * i + 1] * wi;
        a3z += v3[3 * i + 2] * wi;
        a1  += v1[i] * wi;

        sx += sv[3 * i + 0];
        sy += sv[3 * i + 1];
        sz += sv[3 * i + 2];
    }

    bg[r] = expf(run);                       // empty ray: exp(0) = 1, matches reference
    i3[3 * r + 0] = a3x; i3[3 * r + 1] = a3y; i3[3 * r + 2] = a3z;
    i1[r] = a1;
    spr[3 * r + 0] = sx; spr[3 * r + 1] = sy; spr[3 * r + 2] = sz;
}

// -----------------------------------------------------------------------------
// Kernel 2: one thread per sample. Stage the ray_id tile through LDS with the
// CDNA5 async global->LDS path (ASYNCcnt), then gather sum_per_ray and write
// sum_per_sample coalesced.
// -----------------------------------------------------------------------------
__global__ void __launch_bounds__(256)
vr_broadcast(const int*   __restrict__ rid,
             const float* __restrict__ spr,
             float*       __restrict__ out,   // [N,3]
             int n)
{
    __shared__ int tile[256];
    const int i = blockIdx.x * 256 + threadIdx.x;

    if (i < n) {
        // LDS byte address of this lane's slot (low 32 bits of the generic
        // pointer are the wave-relative LDS address on gfx1250).
        unsigned lds_off = (unsigned)(uintptr_t)(&tile[threadIdx.x]);
        unsigned long long gaddr = (unsigned long long)(uintptr_t)(rid + i);
        asm volatile("global_load_async_to_lds_b32 %0, %1, off"
                     :: "v"(lds_off), "v"(gaddr)
                     : "memory");
    }
    asm volatile("s_wait_asynccnt 0x0" ::: "memory");
    __syncthreads();

    if (i < n) {
        const int r = tile[threadIdx.x];
        const float x = spr[3 * r + 0];
        const float y = spr[3 * r + 1];
        const float z = spr[3 * r + 2];
        out[3 * i + 0] = x;
        out[3 * i + 1] = y;
        out[3 * i + 2] = z;
    }
}

// -----------------------------------------------------------------------------
// Host launcher. Output layout (flat, reference return order):
//   [0, N)                transmittance            [N,1]
//   [N, N+R)              bg_transmittance         [R]
//   [N+R, N+4R)           integrated_3d            [R,3]
//   [N+4R, N+5R)          integrated_1d            [R,1]
//   [N+5R, N+8R)          sum_per_ray              [R,3]
//   [N+8R, 4N+8R)         sum_per_sample           [N,3]
// out_size = 4N + 8R  =>  R = (out_size - 4N) / 8
// -----------------------------------------------------------------------------
extern "C" void kernel_launch(void* const* d_in, const int* in_sizes, int n_in,
                              void* d_out, int out_size, void* d_ws, size_t ws_size,
                              hipStream_t stream) {
    (void)n_in; (void)d_ws; (void)ws_size;

    const float* alpha = (const float*)d_in[0];
    const float* v3    = (const float*)d_in[1];
    const float* v1    = (const float*)d_in[2];
    const float* w     = (const float*)d_in[3];
    const float* sv    = (const float*)d_in[4];
    const int*   rid   = (const int*)d_in[5];

    const int n = in_sizes[0];                 // N_SAMPLES (alpha is [N,1])
    const int R = (out_size - 4 * n) / 8;      // N_RAYS

    float* trans = (float*)d_out;
    float* bg    = trans + n;
    float* i3    = bg + R;
    float* i1    = i3 + 3 * R;
    float* spr   = i1 + R;
    float* sps   = spr + 3 * R;

    vr_ray_scan<<<dim3((R + 255) / 256), dim3(256), 0, stream>>>(
        alpha, v3, v1, w, sv, rid, n, R, trans, bg, i3, i1, spr);

    vr_broadcast<<<dim3((n + 255) / 256), dim3(256), 0, stream>>>(
        rid, spr, sps, n);
}